// Predictor_TF_REG_47966194762120
// MI455X (gfx1250) — compile-verified
//
#include <hip/hip_runtime.h>
#include <hip/hip_bf16.h>
#include <math.h>
#include <stdint.h>

// Problem sizes (fixed by the reference)
#define B_SZ  1024
#define IN_SZ 256
#define H_SZ  512
#define P_SZ  128

typedef __attribute__((ext_vector_type(2))) float v2f;
typedef __attribute__((ext_vector_type(8))) float v8f;

__device__ __forceinline__ float gelu_exact(float x) {
    // exact GELU: 0.5*x*(1+erf(x/sqrt(2)))
    return 0.5f * x * (1.0f + erff(x * 0.7071067811865476f));
}

// CDNA5 async global->LDS copy (ASYNCcnt-tracked, no VGPR round trip).
// lds_byte_addr: LDS byte address (low 32 bits of the flat shared pointer,
// per ISA aperture rules). gsrc: 64-bit global address. 16B per lane.
__device__ __forceinline__ void async_copy_b128(uint32_t lds_byte_addr,
                                                const float* gsrc) {
    asm volatile("global_load_async_to_lds_b128 %0, %1, off"
                 :: "v"(lds_byte_addr), "v"(gsrc)
                 : "memory");
}

// ---------------------------------------------------------------------------
// C[M,N] = A[M,K] @ W[N,K]^T + bias            (MODE 0)
// C[M,N] = res  + gelu(A@W^T + bias)           (MODE 1)
// fp32 WMMA (V_WMMA_F32_16X16X4_F32). Workgroup = 256 threads = 8 waves,
// C tile 128x64, wave tile 32x32 (2x2 accumulators), BK = 32.
// Double-buffered LDS tiles filled by GLOBAL_LOAD_ASYNC_TO_LDS_B128; the
// last tile is peeled so the steady-state loop is branch-free: issue 6
// async ops for tile k+1, s_wait_asynccnt<=6 (in-order completion => tile k
// landed), barrier, 32 WMMAs, barrier. Buffer select is pure address math.
// LDS row stride 36 floats: 16B-aligned b128 stores (144B pitch),
// conflict-free b64 fragment reads (36*i mod 64 distinct for i<16).
// ---------------------------------------------------------------------------
template <int MODE>
__global__ __launch_bounds__(256) void gemm_wmma_f32(
    const float* __restrict__ A, const float* __restrict__ W,
    const float* __restrict__ bias, const float* __restrict__ res,
    float* __restrict__ C, int M, int N, int K)
{
    constexpr int BM = 128, BN = 64, BK = 32, LDA = BK + 4, LDW = BK + 4;
    constexpr uint32_t ABYTES = BM * LDA * 4;   // one A buffer, bytes
    constexpr uint32_t WBYTES = BN * LDW * 4;   // one W buffer, bytes
    __shared__ __align__(16) float As[2][BM * LDA];
    __shared__ __align__(16) float Ws[2][BN * LDW];

    const int tid  = threadIdx.x;
    const int wave = tid >> 5;
    const int lane = tid & 31;
    const int half = lane >> 4;   // 0: lanes 0-15, 1: lanes 16-31
    const int lr   = lane & 15;
    const int wy   = wave >> 1;   // 0..3  (M direction)
    const int wx   = wave & 1;    // 0..1  (N direction)
    const int m0   = blockIdx.y * BM;
    const int n0   = blockIdx.x * BN;

    // --- per-thread staging slots: 4 b128 for A, 2 b128 for W --------------
    const float* gA[4];
    uint32_t     lA[4];
    #pragma unroll
    for (int l = 0; l < 4; ++l) {
        const int idx = tid + l * 256;
        const int r = idx >> 3;            // 0..127
        const int c = (idx & 7) << 2;      // 0,4,...,28
        gA[l] = A + (size_t)(m0 + r) * K + c;
        lA[l] = (uint32_t)(uintptr_t)&As[0][r * LDA + c];
    }
    const float* gWp[2];
    uint32_t     lW[2];
    #pragma unroll
    for (int l = 0; l < 2; ++l) {
        const int idx = tid + l * 256;
        const int r = idx >> 3;            // 0..63
        const int c = (idx & 7) << 2;
        gWp[l] = W + (size_t)(n0 + r) * K + c;
        lW[l]  = (uint32_t)(uintptr_t)&Ws[0][r * LDW + c];
    }

    v8f acc[2][2] = {};
    const int nk = K / BK;                 // 8 (K=256) or 16 (K=512)

    // prefetch tile 0 into buffer 0
    #pragma unroll
    for (int l = 0; l < 4; ++l) async_copy_b128(lA[l], gA[l]);
    #pragma unroll
    for (int l = 0; l < 2; ++l) async_copy_b128(lW[l], gWp[l]);
    #pragma unroll
    for (int l = 0; l < 4; ++l) gA[l] += BK;
    #pragma unroll
    for (int l = 0; l < 2; ++l) gWp[l] += BK;

    // --- steady state: branch-free prefetch + compute ----------------------
    for (int kb = 0; kb < nk - 1; ++kb) {
        const uint32_t nb = (uint32_t)((kb + 1) & 1);   // dest buffer for tile kb+1
        #pragma unroll
        for (int l = 0; l < 4; ++l) async_copy_b128(lA[l] + nb * ABYTES, gA[l]);
        #pragma unroll
        for (int l = 0; l < 2; ++l) async_copy_b128(lW[l] + nb * WBYTES, gWp[l]);
        #pragma unroll
        for (int l = 0; l < 4; ++l) gA[l] += BK;
        #pragma unroll
        for (int l = 0; l < 2; ++l) gWp[l] += BK;
        // 6 ops outstanding for tile kb+1; <=6 => tile kb's ops landed
        asm volatile("s_wait_asynccnt 0x6" ::: "memory");
        __syncthreads();                    // all waves' tile-kb data in LDS

        const int buf = kb & 1;
        const float* as = &As[0][0] + buf * (BM * LDA);
        const float* ws = &Ws[0][0] + buf * (BN * LDW);
        #pragma unroll
        for (int ks = 0; ks < BK; ks += 4) {
            const int colf = ks + 2 * half;  // K-pair per half-wave (ISA 16x4 f32 layout)
            v2f a0 = *(const v2f*)(as + (wy * 32 +  0 + lr) * LDA + colf);
            v2f a1 = *(const v2f*)(as + (wy * 32 + 16 + lr) * LDA + colf);
            v2f b0 = *(const v2f*)(ws + (wx * 32 +  0 + lr) * LDW + colf);
            v2f b1 = *(const v2f*)(ws + (wx * 32 + 16 + lr) * LDW + colf);
            acc[0][0] = __builtin_amdgcn_wmma_f32_16x16x4_f32(false, a0, false, b0,
                            (short)0, acc[0][0], false, false);
            acc[0][1] = __builtin_amdgcn_wmma_f32_16x16x4_f32(false, a0, false, b1,
                            (short)0, acc[0][1], false, false);
            acc[1][0] = __builtin_amdgcn_wmma_f32_16x16x4_f32(false, a1, false, b0,
                            (short)0, acc[1][0], false, false);
            acc[1][1] = __builtin_amdgcn_wmma_f32_16x16x4_f32(false, a1, false, b1,
                            (short)0, acc[1][1], false, false);
        }
        __syncthreads();                    // last-reader gate before refill
    }

    // --- peeled final tile -------------------------------------------------
    {
        asm volatile("s_wait_asynccnt 0x0" ::: "memory");
        __syncthreads();
        const int buf = (nk - 1) & 1;
        const float* as = &As[0][0] + buf * (BM * LDA);
        const float* ws = &Ws[0][0] + buf * (BN * LDW);
        #pragma unroll
        for (int ks = 0; ks < BK; ks += 4) {
            const int colf = ks + 2 * half;
            v2f a0 = *(const v2f*)(as + (wy * 32 +  0 + lr) * LDA + colf);
            v2f a1 = *(const v2f*)(as + (wy * 32 + 16 + lr) * LDA + colf);
            v2f b0 = *(const v2f*)(ws + (wx * 32 +  0 + lr) * LDW + colf);
            v2f b1 = *(const v2f*)(ws + (wx * 32 + 16 + lr) * LDW + colf);
            acc[0][0] = __builtin_amdgcn_wmma_f32_16x16x4_f32(false, a0, false, b0,
                            (short)0, acc[0][0], false, false);
            acc[0][1] = __builtin_amdgcn_wmma_f32_16x16x4_f32(false, a0, false, b1,
                            (short)0, acc[0][1], false, false);
            acc[1][0] = __builtin_amdgcn_wmma_f32_16x16x4_f32(false, a1, false, b0,
                            (short)0, acc[1][0], false, false);
            acc[1][1] = __builtin_amdgcn_wmma_f32_16x16x4_f32(false, a1, false, b1,
                            (short)0, acc[1][1], false, false);
        }
    }

    // --- epilogue: C/D layout: VGPR g -> row = g + 8*half, col = lane%16 ----
    #pragma unroll
    for (int mt = 0; mt < 2; ++mt) {
        #pragma unroll
        for (int nt = 0; nt < 2; ++nt) {
            const int col = n0 + wx * 32 + nt * 16 + lr;
            const float bv = bias[col];
            #pragma unroll
            for (int g = 0; g < 8; ++g) {
                const int row = m0 + wy * 32 + mt * 16 + g + 8 * half;
                float vl = acc[mt][nt][g] + bv;
                if (MODE == 1) vl = res[(size_t)row * N + col] + gelu_exact(vl);
                C[(size_t)row * N + col] = vl;
            }
        }
    }
}

// ---------------------------------------------------------------------------
// out[b,i] = x[b,i] + sum_j softmax_j(q[b,i]*k[b,j]) * v[b,j]
// One workgroup per batch row; k,v in LDS; stable max via s*kmax / s*kmin.
// ---------------------------------------------------------------------------
__global__ __launch_bounds__(256) void attn_kernel(
    const float* __restrict__ x, const float* __restrict__ q,
    const float* __restrict__ k, const float* __restrict__ v,
    float* __restrict__ out)
{
    __shared__ float sk[H_SZ];
    __shared__ float sv[H_SZ];
    __shared__ float smax[256];
    __shared__ float smin[256];

    const int b   = blockIdx.x;
    const int tid = threadIdx.x;
    const size_t base = (size_t)b * H_SZ;

    float ka = k[base + tid], kbv = k[base + tid + 256];
    sk[tid] = ka;               sk[tid + 256] = kbv;
    sv[tid] = v[base + tid];    sv[tid + 256] = v[base + tid + 256];
    smax[tid] = fmaxf(ka, kbv);
    smin[tid] = fminf(ka, kbv);
    __syncthreads();
    for (int s = 128; s > 0; s >>= 1) {
        if (tid < s) {
            smax[tid] = fmaxf(smax[tid], smax[tid + s]);
            smin[tid] = fminf(smin[tid], smin[tid + s]);
        }
        __syncthreads();
    }
    const float kmax = smax[0], kmin = smin[0];

    #pragma unroll
    for (int ii = 0; ii < 2; ++ii) {
        const int i = tid + ii * 256;
        const float s = q[base + i];
        const float m = (s >= 0.0f) ? s * kmax : s * kmin;  // max_j s*k_j
        float num = 0.0f, den = 0.0f;
        #pragma unroll 8
        for (int j = 0; j < H_SZ; ++j) {
            float e = __expf(fmaf(s, sk[j], -m));
            num = fmaf(e, sv[j], num);
            den += e;
        }
        out[base + i] = x[base + i] + num / den;
    }
}

// ---------------------------------------------------------------------------
// out[b] = dot(fea[b,:], reg_w) + reg_b     (P = 128, one block per row)
// ---------------------------------------------------------------------------
__global__ __launch_bounds__(128) void reg_kernel(
    const float* __restrict__ fea, const float* __restrict__ rw,
    const float* __restrict__ rb, float* __restrict__ out)
{
    __shared__ float sred[128];
    const int b = blockIdx.x, tid = threadIdx.x;
    sred[tid] = fea[(size_t)b * P_SZ + tid] * rw[tid];
    __syncthreads();
    for (int s = 64; s > 0; s >>= 1) {
        if (tid < s) sred[tid] += sred[tid + s];
        __syncthreads();
    }
    if (tid == 0) out[b] = sred[0] + rb[0];
}

// ---------------------------------------------------------------------------
extern "C" void kernel_launch(void* const* d_in, const int* in_sizes, int n_in,
                              void* d_out, int out_size, void* d_ws, size_t ws_size,
                              hipStream_t stream)
{
    (void)in_sizes; (void)n_in; (void)out_size; (void)ws_size;

    const float* x      = (const float*)d_in[0];
    const float* lin0_w = (const float*)d_in[1];
    const float* lin0_b = (const float*)d_in[2];
    const float* b1_qw  = (const float*)d_in[3];  const float* b1_qb  = (const float*)d_in[4];
    const float* b1_kw  = (const float*)d_in[5];  const float* b1_kb  = (const float*)d_in[6];
    const float* b1_vw  = (const float*)d_in[7];  const float* b1_vb  = (const float*)d_in[8];
    const float* b1_f1w = (const float*)d_in[9];  const float* b1_f1b = (const float*)d_in[10];
    const float* b2_qw  = (const float*)d_in[11]; const float* b2_qb  = (const float*)d_in[12];
    const float* b2_kw  = (const float*)d_in[13]; const float* b2_kb  = (const float*)d_in[14];
    const float* b2_vw  = (const float*)d_in[15]; const float* b2_vb  = (const float*)d_in[16];
    const float* b2_f1w = (const float*)d_in[17]; const float* b2_f1b = (const float*)d_in[18];
    const float* fea_w  = (const float*)d_in[19]; const float* fea_b  = (const float*)d_in[20];
    const float* reg_w  = (const float*)d_in[21]; const float* reg_b  = (const float*)d_in[22];

    float* out = (float*)d_out;          // [B]         (tuple element 0)
    float* fea = (float*)d_out + B_SZ;   // [B,P] flat  (tuple element 1)

    const size_t BH = (size_t)B_SZ * H_SZ;
    float* ws = (float*)d_ws;            // 5 * 2MB fp32 scratch
    float* h  = ws;
    float* qv = ws + 1 * BH;
    float* kv = ws + 2 * BH;
    float* vv = ws + 3 * BH;
    float* ha = ws + 4 * BH;

    const dim3 blk(256);
    const dim3 gH(H_SZ / 64, B_SZ / 128);   // N=512 GEMMs
    const dim3 gP(P_SZ / 64, B_SZ / 128);   // N=128 GEMM

    // lin0: h = x @ lin0_w^T + b
    gemm_wmma_f32<0><<<gH, blk, 0, stream>>>(x, lin0_w, lin0_b, nullptr, h,
                                             B_SZ, H_SZ, IN_SZ);
    // ---- block 1 ----
    gemm_wmma_f32<0><<<gH, blk, 0, stream>>>(h, b1_qw, b1_qb, nullptr, qv, B_SZ, H_SZ, H_SZ);
    gemm_wmma_f32<0><<<gH, blk, 0, stream>>>(h, b1_kw, b1_kb, nullptr, kv, B_SZ, H_SZ, H_SZ);
    gemm_wmma_f32<0><<<gH, blk, 0, stream>>>(h, b1_vw, b1_vb, nullptr, vv, B_SZ, H_SZ, H_SZ);
    attn_kernel<<<B_SZ, 256, 0, stream>>>(h, qv, kv, vv, ha);
    gemm_wmma_f32<1><<<gH, blk, 0, stream>>>(ha, b1_f1w, b1_f1b, ha, h, B_SZ, H_SZ, H_SZ);
    // ---- block 2 ----
    gemm_wmma_f32<0><<<gH, blk, 0, stream>>>(h, b2_qw, b2_qb, nullptr, qv, B_SZ, H_SZ, H_SZ);
    gemm_wmma_f32<0><<<gH, blk, 0, stream>>>(h, b2_kw, b2_kb, nullptr, kv, B_SZ, H_SZ, H_SZ);
    gemm_wmma_f32<0><<<gH, blk, 0, stream>>>(h, b2_vw, b2_vb, nullptr, vv, B_SZ, H_SZ, H_SZ);
    attn_kernel<<<B_SZ, 256, 0, stream>>>(h, qv, kv, vv, ha);
    gemm_wmma_f32<1><<<gH, blk, 0, stream>>>(ha, b2_f1w, b2_f1b, ha, h, B_SZ, H_SZ, H_SZ);
    // ---- heads ----
    gemm_wmma_f32<0><<<gP, blk, 0, stream>>>(h, fea_w, fea_b, nullptr, fea, B_SZ, P_SZ, H_SZ);
    reg_kernel<<<B_SZ, 128, 0, stream>>>(fea, reg_w, reg_b, out);
}